// VanillaGCN_43782896616158
// MI455X (gfx1250) — compile-verified
//
#include <hip/hip_runtime.h>
#include <math.h>

// ---------------------------------------------------------------------------
// 2-layer GCN (PyG GCNConv semantics) for MI455X / gfx1250.
//  - Dense transforms use V_WMMA_F32_16X16X4_F32 (wave32, fp32-exact).
//  - Edge aggregation via global_atomic_add_f32 (accumulators live in 192MB L2).
//  - Self-loop terms folded analytically: contrib = h[n] * dinv[n]^2.
// ---------------------------------------------------------------------------

typedef __attribute__((ext_vector_type(2))) float v2f;
typedef __attribute__((ext_vector_type(8))) float v8f;

#define F_IN  128
#define F_HID 16
#define F_OUT 64

// NOTE: reference declares edge_index as int64, but JAX without x64 emits
// int32 and the harness contract says "integer -> const int*".
typedef int eidx_t;

__device__ __forceinline__ v8f wmma_f32_16x16x4(v2f a, v2f b, v8f c) {
  // 8 args: (neg_a, A, neg_b, B, c_mod, C, reuse_a, reuse_b)
  return __builtin_amdgcn_wmma_f32_16x16x4_f32(false, a, false, b, (short)0, c,
                                               false, false);
}

// ---- init: zero d_out (agg2 target) and agg1, seed dinv with self-loop -----
__global__ void gcn_init(float* __restrict__ out64, float* __restrict__ agg1,
                         float* __restrict__ dinv, int N) {
  int tid = blockIdx.x * blockDim.x + threadIdx.x;
  if (tid < N * F_OUT) out64[tid] = 0.0f;
  if (tid < N * F_HID) agg1[tid] = 0.0f;
  if (tid < N) dinv[tid] = 1.0f;  // self-loop contributes 1 to in-degree
}

// ---- in-degree count over dst ---------------------------------------------
__global__ void gcn_degree(const eidx_t* __restrict__ dst,
                           float* __restrict__ deg, int E) {
  int e = blockIdx.x * blockDim.x + threadIdx.x;
  if (e >= E) return;
  atomicAdd(&deg[dst[e]], 1.0f);
}

// ---- deg -> deg^{-1/2} in place -------------------------------------------
__global__ void gcn_rsqrt(float* __restrict__ dinv, int N) {
  int n = blockIdx.x * blockDim.x + threadIdx.x;
  if (n >= N) return;
  float d = dinv[n];
  dinv[n] = (d > 0.0f) ? rsqrtf(d) : 0.0f;
}

// ---- layer-1 GEMM: h1[N,16] = x[N,128] @ W1[128,16] via WMMA ---------------
__global__ void gcn_gemm1_wmma(const float* __restrict__ x,
                               const float* __restrict__ W1,
                               float* __restrict__ h1, int nTiles) {
  int wave = (blockIdx.x * blockDim.x + threadIdx.x) >> 5;
  if (wave >= nTiles) return;               // wave-uniform: EXEC stays all-1s
  int lane = threadIdx.x & 31;
  int m  = lane & 15;                       // A row / B-D column
  int kh = (lane >> 4) << 1;                // K sub-offset: 0 or 2
  const float* xrow = x + (size_t)(wave * 16 + m) * F_IN;
  v8f acc = {};
#pragma unroll
  for (int kk = 0; kk < F_IN; kk += 4) {
    v2f a, b;
    a.x = xrow[kk + kh];
    a.y = xrow[kk + kh + 1];
    b.x = W1[(kk + kh) * F_HID + m];
    b.y = W1[(kk + kh + 1) * F_HID + m];
    acc = wmma_f32_16x16x4(a, b, acc);
  }
  int mh = (lane >> 4) * 8;                 // D: lanes 16-31 hold M = r+8
#pragma unroll
  for (int r = 0; r < 8; ++r)
    h1[(size_t)(wave * 16 + mh + r) * F_HID + m] = acc[r];
}

// ---- layer-1 edge aggregation: agg1[dst] += h1[src] * norm -----------------
__global__ void gcn_aggregate1(const eidx_t* __restrict__ src,
                               const eidx_t* __restrict__ dst,
                               const float* __restrict__ dinv,
                               const float* __restrict__ h1,
                               float* __restrict__ agg1, int E) {
  int tid = blockIdx.x * blockDim.x + threadIdx.x;
  int e = tid >> 4, f = tid & 15;
  if (e >= E) return;
  int s = src[e], d = dst[e];
  float w = dinv[s] * dinv[d];
  atomicAdd(&agg1[(size_t)d * F_HID + f], h1[(size_t)s * F_HID + f] * w);
}

// ---- t = relu(agg1 + selfloop + b1) ---------------------------------------
__global__ void gcn_relu(const float* __restrict__ agg1,
                         const float* __restrict__ h1,
                         const float* __restrict__ dinv,
                         const float* __restrict__ b1,
                         float* __restrict__ t, int N) {
  int tid = blockIdx.x * blockDim.x + threadIdx.x;
  int n = tid >> 4, f = tid & 15;
  if (n >= N) return;
  float di = dinv[n];
  float v = agg1[tid] + h1[tid] * di * di + b1[f];
  t[tid] = v > 0.0f ? v : 0.0f;
}

// ---- layer-2 GEMM: h2[N,64] = t[N,16] @ W2[16,64] via WMMA -----------------
__global__ void gcn_gemm2_wmma(const float* __restrict__ t,
                               const float* __restrict__ W2,
                               float* __restrict__ h2, int nTiles) {
  int wave = (blockIdx.x * blockDim.x + threadIdx.x) >> 5;
  if (wave >= nTiles) return;               // wave-uniform
  int lane = threadIdx.x & 31;
  int m  = lane & 15;
  int kh = (lane >> 4) << 1;
  const float* trow = t + (size_t)(wave * 16 + m) * F_HID;
  v8f acc[4];
  acc[0] = acc[1] = acc[2] = acc[3] = (v8f){};
#pragma unroll
  for (int kk = 0; kk < F_HID; kk += 4) {
    v2f a;
    a.x = trow[kk + kh];
    a.y = trow[kk + kh + 1];
#pragma unroll
    for (int nt = 0; nt < 4; ++nt) {
      v2f b;
      b.x = W2[(kk + kh) * F_OUT + nt * 16 + m];
      b.y = W2[(kk + kh + 1) * F_OUT + nt * 16 + m];
      acc[nt] = wmma_f32_16x16x4(a, b, acc[nt]);
    }
  }
  int mh = (lane >> 4) * 8;
#pragma unroll
  for (int nt = 0; nt < 4; ++nt)
#pragma unroll
    for (int r = 0; r < 8; ++r)
      h2[(size_t)(wave * 16 + mh + r) * F_OUT + nt * 16 + m] = acc[nt][r];
}

// ---- layer-2 edge aggregation: out[dst] += h2[src] * norm ------------------
__global__ void gcn_aggregate2(const eidx_t* __restrict__ src,
                               const eidx_t* __restrict__ dst,
                               const float* __restrict__ dinv,
                               const float* __restrict__ h2,
                               float* __restrict__ out, int E) {
  int tid = blockIdx.x * blockDim.x + threadIdx.x;
  int e = tid >> 6, f = tid & 63;
  if (e >= E) return;
  int s = src[e], d = dst[e];
  float w = dinv[s] * dinv[d];
  atomicAdd(&out[(size_t)d * F_OUT + f], h2[(size_t)s * F_OUT + f] * w);
}

// ---- finalize: out = log_softmax(out + selfloop + b2), one wave per node ---
__global__ void gcn_finalize(float* __restrict__ out,
                             const float* __restrict__ h2,
                             const float* __restrict__ dinv,
                             const float* __restrict__ b2, int N) {
  int wave = (blockIdx.x * blockDim.x + threadIdx.x) >> 5;
  if (wave >= N) return;
  int lane = threadIdx.x & 31;
  size_t base = (size_t)wave * F_OUT;
  float di = dinv[wave];
  float sl = di * di;
  float v0 = out[base + lane]      + h2[base + lane]      * sl + b2[lane];
  float v1 = out[base + lane + 32] + h2[base + lane + 32] * sl + b2[lane + 32];
  float mx = fmaxf(v0, v1);
#pragma unroll
  for (int off = 16; off > 0; off >>= 1)
    mx = fmaxf(mx, __shfl_xor(mx, off, 32));
  float s = __expf(v0 - mx) + __expf(v1 - mx);
#pragma unroll
  for (int off = 16; off > 0; off >>= 1)
    s += __shfl_xor(s, off, 32);
  float lse = mx + __logf(s);
  out[base + lane]      = v0 - lse;
  out[base + lane + 32] = v1 - lse;
}

extern "C" void kernel_launch(void* const* d_in, const int* in_sizes, int n_in,
                              void* d_out, int out_size, void* d_ws, size_t ws_size,
                              hipStream_t stream) {
  const float*  x  = (const float*)d_in[0];
  const eidx_t* ei = (const eidx_t*)d_in[1];
  const float*  W1 = (const float*)d_in[2];
  const float*  b1 = (const float*)d_in[3];
  const float*  W2 = (const float*)d_in[4];
  const float*  b2 = (const float*)d_in[5];
  float* out = (float*)d_out;

  const int N = in_sizes[0] / F_IN;
  const int E = in_sizes[1] / 2;
  const eidx_t* src = ei;
  const eidx_t* dst = ei + E;

  // workspace carve-up (floats): dinv | h1 | agg1 | t | h2  (~45 MB total)
  float* wsf  = (float*)d_ws;
  float* dinv = wsf;
  float* h1   = dinv + N;
  float* agg1 = h1 + (size_t)N * F_HID;
  float* t    = agg1 + (size_t)N * F_HID;
  float* h2   = t + (size_t)N * F_HID;

  const int B = 256;
  const int nTiles = (N + 15) / 16;
  const int gemmBlocks = (nTiles * 32 + B - 1) / B;

  gcn_init<<<(N * F_OUT + B - 1) / B, B, 0, stream>>>(out, agg1, dinv, N);
  gcn_degree<<<(E + B - 1) / B, B, 0, stream>>>(dst, dinv, E);
  gcn_rsqrt<<<(N + B - 1) / B, B, 0, stream>>>(dinv, N);
  gcn_gemm1_wmma<<<gemmBlocks, B, 0, stream>>>(x, W1, h1, nTiles);
  gcn_aggregate1<<<(E * 16 + B - 1) / B, B, 0, stream>>>(src, dst, dinv, h1, agg1, E);
  gcn_relu<<<(N * F_HID + B - 1) / B, B, 0, stream>>>(agg1, h1, dinv, b1, t, N);
  gcn_gemm2_wmma<<<gemmBlocks, B, 0, stream>>>(t, W2, h2, nTiles);
  gcn_aggregate2<<<(E * 64 + B - 1) / B, B, 0, stream>>>(src, dst, dinv, h2, out, E);
  gcn_finalize<<<(N * 32 + B - 1) / B, B, 0, stream>>>(out, h2, dinv, b2, N);
}